// ResBlock_69870527971810
// MI455X (gfx1250) — compile-verified
//
#include <hip/hip_runtime.h>

#define N_EDGES  100000
#define N_NODES  10000
#define CHANNELS 16
#define HIDDEN   (N_NODES * CHANNELS)
#define NNZ      (N_EDGES * CHANNELS)
#define BATCH    64

typedef __attribute__((ext_vector_type(2))) float v2f;
typedef __attribute__((ext_vector_type(8))) float v8f;

#if __has_builtin(__builtin_amdgcn_global_load_async_to_lds_b32) && \
    __has_builtin(__builtin_amdgcn_s_wait_asynccnt)
#define HAVE_ASYNC_LDS 1
#else
#define HAVE_ASYNC_LDS 0
#endif

// Guaranteed single-instruction f32 atomic add (no-return form, STOREcnt).
// ISA: store-source VGPRs may be overwritten after issue (no hazard); S_ENDPGM
// performs an implicit wait-idle, so outstanding atomics drain at kernel end.
__device__ __forceinline__ void atomAddF32(float* p, float v) {
#if defined(__gfx1250__)
  asm volatile("global_atomic_add_f32 %0, %1, off" ::"v"(p), "v"(v) : "memory");
#else
  atomicAdd(p, v);
#endif
}

#if HAVE_ASYNC_LDS
typedef __attribute__((address_space(1))) int as1_int;
typedef __attribute__((address_space(3))) int as3_int;
__device__ __forceinline__ void asyncLoadB32(const float* g, float* l) {
  __builtin_amdgcn_global_load_async_to_lds_b32((as1_int*)g, (as3_int*)l, 0, 0);
}
#endif

// ---------------------------------------------------------------- init / zero
__global__ void init_hT(float* __restrict__ hT, const float* __restrict__ b_in) {
  int i = blockIdx.x * 256 + threadIdx.x;     // HIDDEN*64 threads exactly
  hT[i] = b_in[i >> 6];
}

__global__ void zero_yT(float* __restrict__ yT) {
  int i = blockIdx.x * 256 + threadIdx.x;     // N_EDGES*64 threads exactly
  yT[i] = 0.0f;
}

// ------------------------------------------------------- x (B,E) -> xT (E,B)
__global__ void transpose_x(const float* __restrict__ x, float* __restrict__ xT) {
  __shared__ float tile[64][65];
  const int tid = threadIdx.x;
  const int tx  = tid & 63;        // edge offset in tile (phase 1)
  const int tq  = tid >> 6;        // 0..3
  const int e0  = blockIdx.x * 64;
  if (e0 + tx < N_EDGES) {
#pragma unroll
    for (int r = 0; r < 16; ++r) {
      int b = tq * 16 + r;
#if HAVE_ASYNC_LDS
      asyncLoadB32(x + (size_t)b * N_EDGES + e0 + tx, &tile[tx][b]);
#else
      tile[tx][b] = x[(size_t)b * N_EDGES + e0 + tx];
#endif
    }
  }
#if HAVE_ASYNC_LDS
  __builtin_amdgcn_s_wait_asynccnt(0);
#endif
  __syncthreads();
#pragma unroll
  for (int r = 0; r < 16; ++r) {
    int eL = tq * 16 + r;
    if (e0 + eL < N_EDGES)
      xT[(size_t)(e0 + eL) * 64 + tx] = tile[eL][tx];
  }
}

// ------------------------------------------- sparse AXPY scatter (both layers)
// one wave per nonzero: 32 lanes x float2 cover the 64-batch row (256 B line)
__global__ void scatter_axpy(const float* __restrict__ srcT,
                             const int*   __restrict__ src,
                             const int*   __restrict__ dst,
                             const float* __restrict__ vals,
                             float* __restrict__ dstT, int nnz) {
  const int lane = threadIdx.x & 31;
  const int wid  = (blockIdx.x << 3) + (threadIdx.x >> 5);
  const int nw   = gridDim.x << 3;
  for (int n = wid; n < nnz; n += nw) {
    const int   s = __builtin_amdgcn_readfirstlane(src[n]);
    const int   d = __builtin_amdgcn_readfirstlane(dst[n]);
    const float v = vals[n];
    const float2 t = ((const float2*)(srcT + (size_t)s * 64))[lane];
    float* op = dstT + (size_t)d * 64 + 2 * lane;
    atomAddF32(op,     t.x * v);
    atomAddF32(op + 1, t.y * v);
  }
}

// --------------------- per-node groupnorm (16 ch) + ELU + mask, WMMA statistics
// one wave per node; node tile = 16 ch x 64 batch, contiguous 4 KB in hT.
// Stats: ones(16x4) [A] x data-chunk(4x16) [B] accumulated over 4 chunks gives
// column sums (over all 16 channels) replicated across rows of the 16x16 D.
__global__ void gnorm_elu_mask(float* __restrict__ hT,
                               const float* __restrict__ gamma,
                               const float* __restrict__ beta,
                               const float* __restrict__ mask) {
  __shared__ float s_mu[8][64];
  __shared__ float s_rs[8][64];
  const int lane = threadIdx.x & 31;
  const int wave = threadIdx.x >> 5;
  const int half = lane >> 4;      // B-operand half: channels +0/+1 vs +2/+3
  const int l    = lane & 15;      // batch within 16-wide group
  for (int node = (blockIdx.x << 3) + wave; node < N_NODES;
       node += (gridDim.x << 3)) {
    float* p = hT + (size_t)node * (CHANNELS * 64);
    const v2f ones = {1.0f, 1.0f};
#pragma unroll
    for (int g = 0; g < 4; ++g) {                 // 4 batch-groups of 16
      v8f cs = {};
      v8f cq = {};
#pragma unroll
      for (int c = 0; c < 4; ++c) {               // 4 channel-chunks of 4 (K=4)
        const int ch = 4 * c + 2 * half;
        const int b  = 16 * g + l;
        float b0 = p[ch * 64 + b];
        float b1 = p[(ch + 1) * 64 + b];
        v2f bv = {b0, b1};
        v2f bq = {b0 * b0, b1 * b1};
        cs = __builtin_amdgcn_wmma_f32_16x16x4_f32(false, ones, false, bv,
                                                   (short)0, cs, false, false);
        cq = __builtin_amdgcn_wmma_f32_16x16x4_f32(false, ones, false, bq,
                                                   (short)0, cq, false, false);
      }
      if (lane < 16) {             // lanes 0..15 hold batch 16g+l in every row
        float mu  = cs[0] * 0.0625f;
        float var = cq[0] * 0.0625f - mu * mu;
        s_mu[wave][16 * g + l] = mu;
        s_rs[wave][16 * g + l] = rsqrtf(var + 1e-5f);
      }
    }
    const float ga = gamma[node];
    const float be = beta[node];
#pragma unroll 4
    for (int i = 0; i < 32; ++i) {                // 1024 elements / 32 lanes
      int idx = i * 32 + lane;
      int ch  = idx >> 6;
      int b   = idx & 63;
      float m = mask[node * CHANNELS + ch];
      float v = p[idx];
      float y = (v - s_mu[wave][b]) * s_rs[wave][b] * ga + be;
      y = y > 0.0f ? y : (__expf(y) - 1.0f);      // ELU
      p[idx] = y * m;
    }
  }
}

// ------------------------- yT (E,B) -> out (B,E), fused + b_out[e] + x[b][e]
__global__ void residual_out(const float* __restrict__ yT,
                             const float* __restrict__ x,
                             const float* __restrict__ b_out,
                             float* __restrict__ out) {
  __shared__ float tile[64][65];
  const int tid = threadIdx.x;
  const int tx  = tid & 63;
  const int tq  = tid >> 6;
  const int e0  = blockIdx.x * 64;
#pragma unroll
  for (int r = 0; r < 16; ++r) {
    int eL = tq * 16 + r;
    if (e0 + eL < N_EDGES) {
#if HAVE_ASYNC_LDS
      asyncLoadB32(yT + (size_t)(e0 + eL) * 64 + tx, &tile[tx][eL]);
#else
      tile[tx][eL] = yT[(size_t)(e0 + eL) * 64 + tx];  // coalesced over batch
#endif
    }
  }
#if HAVE_ASYNC_LDS
  __builtin_amdgcn_s_wait_asynccnt(0);
#endif
  __syncthreads();
  if (e0 + tx < N_EDGES) {
    const float bo = b_out[e0 + tx];
#pragma unroll
    for (int r = 0; r < 16; ++r) {
      int b = tq * 16 + r;
      out[(size_t)b * N_EDGES + e0 + tx] =
          tile[b][tx] + bo + x[(size_t)b * N_EDGES + e0 + tx];
    }
  }
}

// ---------------------------------------------------------------------- launch
extern "C" void kernel_launch(void* const* d_in, const int* in_sizes, int n_in,
                              void* d_out, int out_size, void* d_ws,
                              size_t ws_size, hipStream_t stream) {
  const float* x          = (const float*)d_in[0];
  const float* w_in_vals  = (const float*)d_in[1];
  const float* b_in       = (const float*)d_in[2];
  const float* w_out_vals = (const float*)d_in[3];
  const float* b_out      = (const float*)d_in[4];
  const float* gamma      = (const float*)d_in[5];
  const float* beta       = (const float*)d_in[6];
  const float* node_mask  = (const float*)d_in[7];
  const int*   w_in_src   = (const int*)d_in[8];
  const int*   w_in_dst   = (const int*)d_in[9];
  const int*   w_out_src  = (const int*)d_in[10];
  const int*   w_out_dst  = (const int*)d_in[11];
  float* out = (float*)d_out;

  // workspace: xT/yT aliased (xT dead after layer-1 scatter) | hT
  const size_t XT_BYTES = (size_t)N_EDGES * 64 * sizeof(float);   // 25.6 MB
  float* xT = (float*)d_ws;                                       // also yT
  float* hT = (float*)((char*)d_ws + XT_BYTES);                   // 41.0 MB

  const int e_tiles = (N_EDGES + 63) / 64;                        // 1563

  init_hT<<<(HIDDEN * 64) / 256, 256, 0, stream>>>(hT, b_in);
  transpose_x<<<e_tiles, 256, 0, stream>>>(x, xT);
  scatter_axpy<<<8192, 256, 0, stream>>>(xT, w_in_src, w_in_dst, w_in_vals,
                                         hT, NNZ);
  gnorm_elu_mask<<<N_NODES / 8, 256, 0, stream>>>(hT, gamma, beta, node_mask);
  zero_yT<<<(N_EDGES * 64) / 256, 256, 0, stream>>>(xT);          // xT reused as yT
  scatter_axpy<<<8192, 256, 0, stream>>>(hT, w_out_src, w_out_dst, w_out_vals,
                                         xT, NNZ);
  residual_out<<<e_tiles, 256, 0, stream>>>(xT, x, b_out, out);
}